// Word2ManCBOW_40097814675559
// MI455X (gfx1250) — compile-verified
//
#include <hip/hip_runtime.h>

// ---------------------------------------------------------------------------
// CBOW forward: log_softmax( (xs @ EMBEDM) @ embed2vocab ), fp32 end-to-end.
// B=1024, V=50257, E=128.  xs is n-hot sparse (<= 8 nonzeros/row).
// ---------------------------------------------------------------------------

#define EMB 128
#define NCTX_MAX 16
#define K1_THREADS 256
#define K2_THREADS 256   // 8 waves * 32 lanes
#define K3_THREADS 256

#define MSUB 4           // M-subtiles (of 16 rows) per wave -> 64 rows/block
#define APITCH 132       // LDS pitch: 132 mod 64 banks == 4 -> conflict-free b64

typedef __attribute__((ext_vector_type(2))) float v2f;
typedef __attribute__((ext_vector_type(4))) float v4f;
typedef __attribute__((ext_vector_type(8))) float v8f;

// ---------------------------------------------------------------------------
// Kernel 1: sparse gather.  One block per batch row.  Deterministic:
// per-thread contiguous-chunk counts -> serial prefix -> compacted (idx,w)
// list in LDS -> 128 threads each own one embedding dim.
// ---------------------------------------------------------------------------
__global__ __launch_bounds__(K1_THREADS)
void cbow_gather_kernel(const float* __restrict__ xs,
                        const float* __restrict__ embedm,
                        float* __restrict__ xsembeds,
                        int Vdim) {
  __shared__ int   s_cnt[K1_THREADS];
  __shared__ int   s_off[K1_THREADS];
  __shared__ int   s_idx[NCTX_MAX];
  __shared__ float s_w[NCTX_MAX];
  __shared__ int   s_n;

  const int b   = blockIdx.x;
  const int tid = threadIdx.x;
  const float* row = xs + (size_t)b * (size_t)Vdim;

  const int chunk = (Vdim + K1_THREADS - 1) / K1_THREADS;
  const int begin = tid * chunk;
  const int end   = (begin + chunk < Vdim) ? (begin + chunk) : Vdim;

  int cnt = 0;
  for (int i = begin; i < end; ++i) cnt += (row[i] != 0.0f) ? 1 : 0;
  s_cnt[tid] = cnt;
  __syncthreads();

  if (tid == 0) {  // tiny serial prefix sum (nnz <= 8, cost negligible)
    int acc = 0;
    for (int t = 0; t < K1_THREADS; ++t) { s_off[t] = acc; acc += s_cnt[t]; }
    s_n = (acc > NCTX_MAX) ? NCTX_MAX : acc;
  }
  __syncthreads();

  int o = s_off[tid];
  for (int i = begin; i < end; ++i) {
    float w = row[i];
    if (w != 0.0f) {
      if (o < NCTX_MAX) { s_idx[o] = i; s_w[o] = w; }
      ++o;
    }
  }
  __syncthreads();

  const int n = s_n;
  if (tid < EMB) {
    float acc = 0.0f;
    for (int j = 0; j < n; ++j)
      acc += s_w[j] * embedm[(size_t)s_idx[j] * EMB + tid];
    xsembeds[(size_t)b * EMB + tid] = acc;
  }
}

// ---------------------------------------------------------------------------
// Kernel 2: logits = xsembeds[1024,128] @ e2v[128,V] via V_WMMA_F32_16X16X4_F32.
// Grid: (ceil(V/128), B/64).  Block = 8 waves; each wave owns a 64x16 output
// strip (4 stacked 16x16 tiles) so every B fragment feeds 4 WMMAs.
//
// Fragment layouts per ISA 7.12.2 (f32, wave32):
//   A 16x4:  lane L -> M = L%16; VGPR0 = A[M][k+2*(L/16)], VGPR1 = +1
//   B 4x16:  lane L -> N = L%16; VGPR0 = B[k+2*(L/16)][N], VGPR1 = +1
//   C/D:     VGPR j -> M = j + 8*(L/16), N = L%16
//
// Out-of-range columns are CLAMPED on load (column N of D depends only on
// column N of B, so duplicated data is harmless) and masked on store.
// ---------------------------------------------------------------------------
__global__ __launch_bounds__(K2_THREADS)
void gemm_wmma_kernel(const float* __restrict__ A,     // [B][128] row-major
                      const float* __restrict__ Bm,    // [128][V] row-major
                      float* __restrict__ out,         // [B][V]
                      int Vdim) {
  __shared__ float As[64 * APITCH];  // 64 rows, padded pitch -> ~33 KB

  const int tid  = threadIdx.x;
  const int row0 = blockIdx.y * 64;

  // Stage the 64x128 A tile, float4 per thread per step, padded LDS pitch.
  {
    const v4f* ag = (const v4f*)(A + (size_t)row0 * EMB);
    #pragma unroll
    for (int it = 0; it < (64 * EMB) / (K2_THREADS * 4); ++it) {
      const int v = it * K2_THREADS + tid;      // which float4 of the tile
      const int r = (v * 4) / EMB;
      const int cidx = (v * 4) % EMB;
      v4f d = ag[v];
      As[r * APITCH + cidx + 0] = d[0];
      As[r * APITCH + cidx + 1] = d[1];
      As[r * APITCH + cidx + 2] = d[2];
      As[r * APITCH + cidx + 3] = d[3];
    }
  }
  __syncthreads();

  const int wave = tid >> 5;
  const int lane = tid & 31;
  const int nIdx = lane & 15;   // N within tile; also M within A subtile
  const int half = lane >> 4;   // selects K sub-pair (+0 or +2)

  const int col  = blockIdx.x * 128 + wave * 16 + nIdx;
  const bool inb = (col < Vdim);
  const int colc = inb ? col : (Vdim - 1);          // clamp: no predication
  const size_t Vs = (size_t)Vdim;

  // B stream pointer: rows (2*half + {0,1}) of k-group, bumped by 4*V each step.
  const float* p = Bm + colc + (size_t)(2 * half) * Vs;

  v8f c0 = {}, c1 = {}, c2 = {}, c3 = {};

  float b0 = p[0];
  float b1 = p[Vs];

  #pragma unroll 4
  for (int k = 0; k < EMB; k += 4) {
    // prefetch next k-step's B pair (uniform branch, scalar-cheap)
    float nb0 = 0.0f, nb1 = 0.0f;
    if (k + 4 < EMB) { nb0 = p[4 * Vs]; nb1 = p[5 * Vs]; }

    v2f bf; bf[0] = b0; bf[1] = b1;

    const int kb = k + 2 * half;
    const int abase = nIdx * APITCH + kb;
    v2f a0, a1, a2, a3;
    a0[0] = As[abase + 0 * 16 * APITCH]; a0[1] = As[abase + 0 * 16 * APITCH + 1];
    a1[0] = As[abase + 1 * 16 * APITCH]; a1[1] = As[abase + 1 * 16 * APITCH + 1];
    a2[0] = As[abase + 2 * 16 * APITCH]; a2[1] = As[abase + 2 * 16 * APITCH + 1];
    a3[0] = As[abase + 3 * 16 * APITCH]; a3[1] = As[abase + 3 * 16 * APITCH + 1];

    // 8 args: (neg_a, A, neg_b, B, c_mod, C, reuse_a, reuse_b)
    c0 = __builtin_amdgcn_wmma_f32_16x16x4_f32(false, a0, false, bf, (short)0, c0, false, false);
    c1 = __builtin_amdgcn_wmma_f32_16x16x4_f32(false, a1, false, bf, (short)0, c1, false, false);
    c2 = __builtin_amdgcn_wmma_f32_16x16x4_f32(false, a2, false, bf, (short)0, c2, false, false);
    c3 = __builtin_amdgcn_wmma_f32_16x16x4_f32(false, a3, false, bf, (short)0, c3, false, false);

    b0 = nb0; b1 = nb1;
    p += 4 * Vs;
  }

  if (inb) {
    float* o0 = out + (size_t)(row0 + 8 * half) * Vs + col;
    #pragma unroll
    for (int j = 0; j < 8; ++j) {
      o0[(size_t)(0 * 16 + j) * Vs] = c0[j];
      o0[(size_t)(1 * 16 + j) * Vs] = c1[j];
      o0[(size_t)(2 * 16 + j) * Vs] = c2[j];
      o0[(size_t)(3 * 16 + j) * Vs] = c3[j];
    }
  }
}

// ---------------------------------------------------------------------------
// Kernel 3: in-place row-wise log_softmax.  One block per row; row re-reads
// are temporally local so they hit L2.  Deterministic tree reductions.
// ---------------------------------------------------------------------------
__global__ __launch_bounds__(K3_THREADS)
void logsoftmax_kernel(float* __restrict__ out, int Vdim) {
  __shared__ float red[K3_THREADS];
  const int b   = blockIdx.x;
  const int tid = threadIdx.x;
  float* row = out + (size_t)b * (size_t)Vdim;

  // pass 1: row max
  float mx = -__builtin_inff();
  for (int i = tid; i < Vdim; i += K3_THREADS) mx = fmaxf(mx, row[i]);
  red[tid] = mx;
  __syncthreads();
  for (int s = K3_THREADS / 2; s > 0; s >>= 1) {
    if (tid < s) red[tid] = fmaxf(red[tid], red[tid + s]);
    __syncthreads();
  }
  mx = red[0];
  __syncthreads();

  // pass 2: sum of exp(x - max)
  float sum = 0.0f;
  for (int i = tid; i < Vdim; i += K3_THREADS) sum += expf(row[i] - mx);
  red[tid] = sum;
  __syncthreads();
  for (int s = K3_THREADS / 2; s > 0; s >>= 1) {
    if (tid < s) red[tid] += red[tid + s];
    __syncthreads();
  }
  const float lse = mx + logf(red[0]);
  __syncthreads();

  // pass 3: normalize in place
  for (int i = tid; i < Vdim; i += K3_THREADS) row[i] -= lse;
}

// ---------------------------------------------------------------------------
// Launcher.  Inputs: [0] xs (B*V f32), [1] metric (E*E, unused),
// [2] EMBEDM (V*E f32), [3] embed2vocab (E*V f32).  Output: B*V f32.
// Workspace: xsembeds, B*E*4 = 512 KB.
// ---------------------------------------------------------------------------
extern "C" void kernel_launch(void* const* d_in, const int* in_sizes, int n_in,
                              void* d_out, int out_size, void* d_ws, size_t ws_size,
                              hipStream_t stream) {
  const float* xs     = (const float*)d_in[0];
  const float* embedm = (const float*)d_in[2];
  const float* e2v    = (const float*)d_in[3];
  float* out = (float*)d_out;
  float* xsembeds = (float*)d_ws;

  const int E_ = EMB;
  const int V_ = in_sizes[2] / E_;        // 50257
  const int B_ = in_sizes[0] / V_;        // 1024

  // K1: sparse bag-of-words gather -> xsembeds [B, 128]
  cbow_gather_kernel<<<B_, K1_THREADS, 0, stream>>>(xs, embedm, xsembeds, V_);

  // K2: WMMA GEMM -> raw logits in d_out (64 rows x 128 cols per block)
  dim3 g2((V_ + 127) / 128, B_ / 64);
  gemm_wmma_kernel<<<g2, K2_THREADS, 0, stream>>>(xsembeds, e2v, out, V_);

  // K3: in-place log_softmax per row
  logsoftmax_kernel<<<B_, K3_THREADS, 0, stream>>>(out, V_);
}